// Sampler_55731495632978
// MI455X (gfx1250) — compile-verified
//
#include <hip/hip_runtime.h>
#include <hip/hip_bf16.h>
#include <stdint.h>

#define BATCH    128
#define VOCAB    128000
#define NTHREADS 512
#define CHUNK    8000          // floats staged per TDM op (32 KB)
#define NCHUNK   (VOCAB / CHUNK)   // 16
#define NBINS    4096
#define CAP      512           // candidate buffer (>= 64 guaranteed by threshold)
#define TOPW     64            // top-k window (top_ks < 64)

#if defined(__has_builtin)
# if __has_builtin(__builtin_amdgcn_tensor_load_to_lds)
#  define HAVE_TDM 1
# endif
# if __has_builtin(__builtin_amdgcn_s_wait_tensorcnt)
#  define HAVE_TWAIT 1
# endif
#endif
#ifndef HAVE_TDM
# define HAVE_TDM 0
#endif
#ifndef HAVE_TWAIT
# define HAVE_TWAIT 0
#endif

typedef unsigned int u32x4 __attribute__((ext_vector_type(4)));
typedef int          i32x8 __attribute__((ext_vector_type(8)));
typedef int          i32x4 __attribute__((ext_vector_type(4)));

__device__ __forceinline__ void wait_tensor_le1() {
#if HAVE_TWAIT
    __builtin_amdgcn_s_wait_tensorcnt(1);
#else
    asm volatile("s_wait_tensorcnt 1" ::: "memory");
#endif
}
__device__ __forceinline__ void wait_tensor_le0() {
#if HAVE_TWAIT
    __builtin_amdgcn_s_wait_tensorcnt(0);
#else
    asm volatile("s_wait_tensorcnt 0" ::: "memory");
#endif
}

#if HAVE_TDM
// Issue one TDM DMA: CHUNK contiguous f32 from global -> LDS (1-D tile).
// D# layout per CDNA5 ISA. EXEC is ignored by TENSOR
// ops; duplicate issue by multiple waves is benign (identical data/dest).
__device__ __forceinline__ void tdm_issue(const float* __restrict__ gsrc,
                                          float* stage) {
    uint64_t ga = (uint64_t)(uintptr_t)gsrc;
    uint32_t lo = (uint32_t)(uintptr_t)stage;   // low 32 bits == LDS byte offset
    u32x4 g0;
    g0[0] = 1u;                                  // count=1, is_restore=0, gather off
    g0[1] = lo;                                  // lds_addr [63:32]
    g0[2] = (uint32_t)ga;                        // global_addr [95:64]
    g0[3] = (uint32_t)((ga >> 32) & 0x01FFFFFFu) // global_addr [120:96]
          | (2u << 30);                          // type=2 ("image")
    i32x8 g1;
    g1[0] = (int)(2u << 16);                               // data_size=4B
    g1[1] = (int)(((unsigned)CHUNK & 0xFFFFu) << 16);      // tensor_dim0[15:0]
    g1[2] = (int)(((unsigned)CHUNK >> 16) | (1u << 16));   // tensor_dim0[31:16], tensor_dim1=1
    g1[3] = (int)(((unsigned)CHUNK & 0xFFFFu) << 16);      // tile_dim0 = CHUNK (1-D tile)
    g1[4] = 0;                                             // tile_dim1/2 unused
    g1[5] = (int)CHUNK;                                    // tensor_dim0_stride[31:0]
    g1[6] = 0;
    g1[7] = 0;
    i32x4 z4; z4[0] = 0; z4[1] = 0; z4[2] = 0; z4[3] = 0;  // groups 2/3 unused (<=2D)
    i32x8 z8;
    z8[0] = 0; z8[1] = 0; z8[2] = 0; z8[3] = 0;
    z8[4] = 0; z8[5] = 0; z8[6] = 0; z8[7] = 0;
    __builtin_amdgcn_tensor_load_to_lds(g0, g1, z4, z4, z8, 0);
}
#endif

// Stream the whole row through double-buffered LDS staging; calls
// consume(stagePtr, chunkBaseIndex) for each chunk. Uniform control flow.
template <typename F>
__device__ __forceinline__ void stream_row(const float* __restrict__ rowp,
                                           float* buf0, float* buf1,
                                           unsigned tid, F&& consume) {
#if HAVE_TDM
    if (tid < 32u) tdm_issue(rowp, buf0);
    #pragma clang loop unroll(disable)
    for (int c = 0; c < NCHUNK; ++c) {
        float* cur = (c & 1) ? buf1 : buf0;
        float* nxt = (c & 1) ? buf0 : buf1;
        if (tid < 32u) {
            if (c + 1 < NCHUNK) {
                tdm_issue(rowp + (size_t)(c + 1) * CHUNK, nxt);
                wait_tensor_le1();     // chunk c complete (in-order completion)
            } else {
                wait_tensor_le0();
            }
        }
        __syncthreads();
        consume(cur, c * CHUNK);
        __syncthreads();
    }
#else
    #pragma clang loop unroll(disable)
    for (int c = 0; c < NCHUNK; ++c) {
        for (unsigned i = tid; i < CHUNK; i += NTHREADS) {
            __builtin_prefetch(&rowp[(size_t)c * CHUNK + i + NTHREADS], 0, 0);
            buf0[i] = rowp[(size_t)c * CHUNK + i];
        }
        __syncthreads();
        consume(buf0, c * CHUNK);
        __syncthreads();
    }
#endif
}

__device__ __forceinline__ uint64_t splitmix64(uint64_t x) {
    x += 0x9E3779B97F4A7C15ULL;
    x ^= x >> 30; x *= 0xBF58476D1CE4E5B9ULL;
    x ^= x >> 27; x *= 0x94D049BB133111EBULL;
    x ^= x >> 31;
    return x;
}

__global__ __launch_bounds__(NTHREADS)
void Sampler_55731495632978_kernel(const float* __restrict__ logits,
                                   const float* __restrict__ temps,
                                   const int*   __restrict__ topks,
                                   const float* __restrict__ topps,
                                   const float* __restrict__ minps,
                                   int* __restrict__ out) {
    __shared__ float s_stageA[CHUNK];
    __shared__ float s_stageB[CHUNK];
    __shared__ int   s_bins[NBINS];
    __shared__ float s_redmax[NTHREADS];
    __shared__ float s_redmin[NTHREADS];
    __shared__ float s_cval[CAP];
    __shared__ int   s_cidx[CAP];
    __shared__ int   s_cnt;
    __shared__ int   s_bstar;
    __shared__ float s_mn, s_scale;

    const int      row  = blockIdx.x;
    const unsigned tid  = threadIdx.x;
    const float    invT = 1.0f / temps[row];
    const int      kk   = topks[row];
    const float    pp   = topps[row];
    const float    mp   = minps[row];
    const float* __restrict__ rowp = logits + (size_t)row * VOCAB;

    // ---- Pass 1: row max/min of scaled values (HBM-bound) ----
    float lmax = -3.0e38f, lmin = 3.0e38f;
    stream_row(rowp, s_stageA, s_stageB, tid,
        [&](const float* st, int /*base*/) {
            for (unsigned i = tid; i < CHUNK; i += NTHREADS) {
                float x = st[i] * invT;
                lmax = fmaxf(lmax, x);
                lmin = fminf(lmin, x);
            }
        });
    s_redmax[tid] = lmax; s_redmin[tid] = lmin;
    __syncthreads();
    #pragma clang loop unroll(disable)
    for (int s = NTHREADS / 2; s > 0; s >>= 1) {
        if (tid < (unsigned)s) {
            s_redmax[tid] = fmaxf(s_redmax[tid], s_redmax[tid + s]);
            s_redmin[tid] = fminf(s_redmin[tid], s_redmin[tid + s]);
        }
        __syncthreads();
    }
    if (tid == 0) {
        float mx = s_redmax[0], mn = s_redmin[0];
        float span = mx - mn;
        if (!(span > 0.0f)) span = 1.0f;
        s_mn = mn;
        s_scale = (float)NBINS / (span * (1.0f + 1e-6f));
    }
    for (unsigned i = tid; i < NBINS; i += NTHREADS) s_bins[i] = 0;
    __syncthreads();
    const float mnv = s_mn, scv = s_scale;

    // ---- Pass 2: histogram (L2-resident) ----
    stream_row(rowp, s_stageA, s_stageB, tid,
        [&](const float* st, int /*base*/) {
            for (unsigned i = tid; i < CHUNK; i += NTHREADS) {
                float x = st[i] * invT;
                int b = (int)((x - mnv) * scv);
                b = b < 0 ? 0 : (b > NBINS - 1 ? NBINS - 1 : b);
                atomicAdd(&s_bins[b], 1);
            }
        });
    if (tid == 0) {
        int cum = 0, b = NBINS - 1;
        #pragma clang loop unroll(disable)
        for (; b >= 0; --b) { cum += s_bins[b]; if (cum >= TOPW) break; }
        s_bstar = (b < 0) ? 0 : b;
        s_cnt = 0;
    }
    __syncthreads();
    const int bstar = s_bstar;

    // ---- Pass 3: collect candidates with bin >= bstar (count >= 64 by construction) ----
    stream_row(rowp, s_stageA, s_stageB, tid,
        [&](const float* st, int base) {
            for (unsigned i = tid; i < CHUNK; i += NTHREADS) {
                float x = st[i] * invT;
                int b = (int)((x - mnv) * scv);
                b = b < 0 ? 0 : (b > NBINS - 1 ? NBINS - 1 : b);
                if (b >= bstar) {
                    int pos = atomicAdd(&s_cnt, 1);
                    if (pos < CAP) { s_cval[pos] = x; s_cidx[pos] = base + (int)i; }
                }
            }
        });
    int cnt = s_cnt; if (cnt > CAP) cnt = CAP;
    for (unsigned i = tid; i < CAP; i += NTHREADS) {
        if ((int)i >= cnt) { s_cval[i] = -3.0e38f; s_cidx[i] = 0x7FFFFFFF; }
    }
    __syncthreads();

    // ---- Bitonic sort CAP entries: value desc, index asc (deterministic total order) ----
    #pragma clang loop unroll(disable)
    for (int k = 2; k <= CAP; k <<= 1) {
        #pragma clang loop unroll(disable)
        for (int j = k >> 1; j > 0; j >>= 1) {
            for (unsigned i = tid; i < CAP; i += NTHREADS) {
                unsigned ixj = i ^ (unsigned)j;
                if (ixj > i) {
                    float v1 = s_cval[i], v2 = s_cval[ixj];
                    int   d1 = s_cidx[i], d2 = s_cidx[ixj];
                    bool less = (v1 < v2) || (v1 == v2 && d1 > d2);
                    bool descBlock = ((i & (unsigned)k) == 0);
                    if (descBlock == less) {
                        s_cval[i] = v2; s_cval[ixj] = v1;
                        s_cidx[i] = d2; s_cidx[ixj] = d1;
                    }
                }
            }
            __syncthreads();
        }
    }

    // ---- Masks + categorical sampling (thread 0; 64-entry serial work) ----
    if (tid == 0) {
        bool keep[TOPW];
        float cum = 0.0f;                 // exclusive cumsum of ORIGINAL sorted values
        #pragma clang loop unroll(disable)
        for (int r = 0; r < TOPW; ++r) {
            float v = s_cval[r];
            keep[r] = (r < kk) && !(cum > pp);   // top-k then top-p (exclusive-cum > p)
            cum += v;
        }
        float thr = s_cval[0] * mp;              // rank 0 always survives k>=1, p>0
        #pragma clang loop unroll(disable)
        for (int r = 0; r < TOPW; ++r)
            if (keep[r] && s_cval[r] < thr) keep[r] = false;

        float M = 0.0f;                          // max over kept values and the 0-tail
        #pragma clang loop unroll(disable)
        for (int r = 0; r < TOPW; ++r) if (keep[r]) M = fmaxf(M, s_cval[r]);
        float Wk = 0.0f; int keptCount = 0;
        #pragma clang loop unroll(disable)
        for (int r = 0; r < TOPW; ++r)
            if (keep[r]) { Wk += __expf(s_cval[r] - M); ++keptCount; }
        int   zw  = TOPW - keptCount;            // zeroed slots inside the top-64 window
        float wz1 = __expf(-M);                  // weight of each zero entry
        float Wz  = (float)((long long)VOCAB - keptCount) * wz1;

        uint64_t h  = splitmix64(42ULL * 0x9E3779B97F4A7C15ULL ^ (uint64_t)(row + 1));
        float u1 = ((float)((h >> 40) & 0xFFFFFFu) + 0.5f) * (1.0f / 16777216.0f);
        float t  = u1 * (Wk + Wz);

        int token = -1;
        #pragma clang loop unroll(disable)
        for (int r = 0; r < TOPW; ++r) {
            if (token < 0 && keep[r]) {
                t -= __expf(s_cval[r] - M);
                if (t <= 0.0f) token = s_cidx[r];
            }
        }
        if (token < 0) {
            // Landed in the zero-weight region.
            float z = t / wz1; if (z < 0.0f) z = 0.0f;
            if (z < (float)zw) {
                int jz = (int)z; if (jz >= zw) jz = zw - 1;
                int cseen = 0;
                #pragma clang loop unroll(disable)
                for (int r = 0; r < TOPW; ++r) {
                    if (token < 0 && !keep[r]) {
                        if (cseen == jz) token = s_cidx[r];
                        ++cseen;
                    }
                }
            } else {
                // Uniform over tokens outside the top-64 window.
                uint64_t h2 = splitmix64(h ^ 0xD1B54A32D192ED03ULL);
                unsigned t0 = (unsigned)(h2 % (uint64_t)VOCAB);
                #pragma clang loop unroll(disable)
                for (int tries = 0; tries < 70; ++tries) {
                    bool inTop = false;
                    #pragma clang loop unroll(disable)
                    for (int r = 0; r < TOPW; ++r)
                        if (s_cidx[r] == (int)t0) { inTop = true; break; }
                    if (!inTop) break;
                    t0 = (t0 + 1u) % VOCAB;
                }
                token = (int)t0;
            }
        }
        if (token < 0) token = s_cidx[0];
        out[row] = token;
    }
}

extern "C" void kernel_launch(void* const* d_in, const int* in_sizes, int n_in,
                              void* d_out, int out_size, void* d_ws, size_t ws_size,
                              hipStream_t stream) {
    (void)in_sizes; (void)n_in; (void)out_size; (void)d_ws; (void)ws_size;
    const float* logits = (const float*)d_in[0];
    const float* temps  = (const float*)d_in[1];
    const int*   topks  = (const int*)d_in[2];
    const float* topps  = (const float*)d_in[3];
    const float* minps  = (const float*)d_in[4];
    int* out = (int*)d_out;
    hipLaunchKernelGGL(Sampler_55731495632978_kernel,
                       dim3(BATCH), dim3(NTHREADS), 0, stream,
                       logits, temps, topks, topps, minps, out);
}